// HardConstrainedMLP_23888608101259
// MI455X (gfx1250) — compile-verified
//
#include <hip/hip_runtime.h>
#include <math.h>

// ---------------------------------------------------------------------------
// HardConstrainedMLP on MI455X (gfx1250, wave32, WMMA).
//   zk = s @ P + q     with P = I - Aaug^T @ Aaug_inv^T  (512x512 padded)
//   q  = bv @ Aaug_inv^T, y = MLP(concat(bv,cv))
// 1000-iteration scan runs entirely in LDS per 16-row block using
// V_WMMA_F32_16X16X4_F32; P is pre-swizzled into B-fragment lane order so
// each B operand is one global_load_b64 from L2-resident memory, and the
// inner loop is software-pipelined one k-chunk ahead.
// ---------------------------------------------------------------------------

typedef float v2f __attribute__((ext_vector_type(2)));
typedef float v8f __attribute__((ext_vector_type(8)));

#define DPAD   512
#define MDIM   250
#define NDIM   250
#define DREAL  500
#define BATCH  1024
#define N_ITER 1000
#define OMEGA_C  1.8f
#define TWO_SIG  0.2f           // 2*sigma
#define INV_12S  (1.0f/1.2f)    // 1/(1+2*sigma)

__device__ __forceinline__ int iclamp(int v, int hi) { return v < hi ? v : hi; }

// mode: 0 = bias+relu, 1 = bias only, 2 = I - acc (Pmat), 3 = plain
// Branch-free bounds handling: clamped unconditional loads + select-zero.
__global__ __launch_bounds__(256) void gemm_wmma_generic(
    const float* __restrict__ A, int as_m, int as_k, int Ma, int Ka,
    const float* __restrict__ W, int ws_k, int ws_n, int Na,
    const float* __restrict__ bias, float* __restrict__ C,
    int Mout, int mode)
{
    const int lane = threadIdx.x & 31;
    const int wave = threadIdx.x >> 5;
    const int hi   = lane >> 4;          // half-wave select
    const int lm   = lane & 15;
    const int m    = blockIdx.x * 16 + lm;
    const int ksel = hi * 2;
    const int mc   = iclamp(m, Ma - 1);
    const bool mok = (m < Ma);

    v8f acc[4] = {};

    for (int k0 = 0; k0 < Ka; k0 += 4) {
        const int ka  = k0 + ksel;
        const int k0c = iclamp(ka,     Ka - 1);
        const int k1c = iclamp(ka + 1, Ka - 1);
        const bool k0ok = (ka     < Ka);
        const bool k1ok = (ka + 1 < Ka);

        float ax = A[(size_t)mc * as_m + (size_t)k0c * as_k];
        float ay = A[(size_t)mc * as_m + (size_t)k1c * as_k];
        v2f a;
        a.x = (mok && k0ok) ? ax : 0.0f;
        a.y = (mok && k1ok) ? ay : 0.0f;

        #pragma unroll
        for (int t = 0; t < 4; ++t) {
            const int n  = (wave + t * 8) * 16 + lm;
            const int nc = iclamp(n, Na - 1);
            const bool nok = (n < Na);
            float bx = W[(size_t)k0c * ws_k + (size_t)nc * ws_n];
            float by = W[(size_t)k1c * ws_k + (size_t)nc * ws_n];
            v2f b;
            b.x = (k0ok && nok) ? bx : 0.0f;
            b.y = (k1ok && nok) ? by : 0.0f;
            acc[t] = __builtin_amdgcn_wmma_f32_16x16x4_f32(
                false, a, false, b, (short)0, acc[t], false, false);
        }
    }

    #pragma unroll
    for (int t = 0; t < 4; ++t) {
        const int col = (wave + t * 8) * 16 + lm;
        const float bb = (bias != nullptr && col < Na) ? bias[iclamp(col, Na - 1)] : 0.0f;
        #pragma unroll
        for (int v = 0; v < 8; ++v) {
            const int row = blockIdx.x * 16 + v + hi * 8;
            const float a_ = acc[t][v];
            float val;
            if (mode == 2) {
                val = ((row == col) ? 1.0f : 0.0f) - a_;
            } else {
                val = a_ + bb;
                if (mode == 0) val = fmaxf(val, 0.0f);
            }
            if (row >= Mout || col >= Na) val = 0.0f;
            C[(size_t)row * DPAD + col] = val;
        }
    }
}

// Pack x = concat(bv, cv), zero-padded to 512 columns.
__global__ void pack_x_kernel(const float* __restrict__ b,
                              const float* __restrict__ c,
                              float* __restrict__ x)
{
    int idx = blockIdx.x * blockDim.x + threadIdx.x;
    if (idx >= BATCH * DPAD) return;
    int i = idx >> 9, j = idx & (DPAD - 1);
    float v = 0.0f;
    if (j < MDIM)       v = b[i * MDIM + j];
    else if (j < DREAL) v = c[i * NDIM + (j - MDIM)];
    x[idx] = v;
}

// Swizzle P (512x512 row-major) into WMMA B-fragment order:
// dst[((kc*32+ct)*32 + lane)*2 + v] = P[4*kc + 2*(lane>=16) + v][ct*16 + (lane&15)]
__global__ void swizzle_pmat_kernel(const float* __restrict__ Pin,
                                    float* __restrict__ Pout)
{
    int idx = blockIdx.x * blockDim.x + threadIdx.x;
    if (idx >= DPAD * DPAD) return;
    int k = idx >> 9, j = idx & (DPAD - 1);
    int kc   = k >> 2;
    int lane = ((k & 2) ? 16 : 0) + (j & 15);
    int v    = k & 1;
    int ct   = j >> 4;
    Pout[((((kc * 32 + ct) * 32) + lane) << 1) + v] = Pin[idx];
}

// Main solver: one workgroup owns 16 batch rows; runs all 1000 iterations
// with s resident in LDS. 8 waves x 4 column tiles cover 512 columns.
__global__ __launch_bounds__(256) void solver_kernel(
    const float* __restrict__ Psw,   // pre-swizzled P, 512*512
    const float* __restrict__ q,     // 1024x512
    const float* __restrict__ y,     // 1024x512
    float* __restrict__ out)         // 1024x500
{
    __shared__ float s_s[16 * DPAD];
    __shared__ float s_q[16 * DPAD];
    __shared__ float s_y[16 * DPAD];
    __shared__ float s_norm[16];
    __shared__ float s_t[16];
    __shared__ float s_us[16];
    __shared__ float s_tn[16];

    const int tid    = threadIdx.x;
    const int lane   = tid & 31;
    const int wave   = tid >> 5;
    const int hi     = lane >> 4;
    const int lm     = lane & 15;
    const int rowblk = blockIdx.x * 16;

    for (int i = tid; i < 16 * DPAD; i += 256) {
        int r = i >> 9, cc = i & (DPAD - 1);
        s_q[i] = q[(size_t)(rowblk + r) * DPAD + cc];
        s_y[i] = y[(size_t)(rowblk + r) * DPAD + cc];
        s_s[i] = 0.0f;
    }
    if (tid < 16) s_norm[tid] = 0.0f;
    __syncthreads();

    const float* arow  = s_s + lm * DPAD + hi * 2;   // A frag base (LDS)
    // Per-tile B fragment base pointers (global, L2-resident):
    //   frag(kc, t) at Psw + ((kc*32 + wave + t*8)*32 + lane)*2
    const float* pb0 = Psw + (size_t)((wave +  0) * 32 + lane) * 2;
    const float* pb1 = Psw + (size_t)((wave +  8) * 32 + lane) * 2;
    const float* pb2 = Psw + (size_t)((wave + 16) * 32 + lane) * 2;
    const float* pb3 = Psw + (size_t)((wave + 24) * 32 + lane) * 2;
    const size_t kcstep = 32 * 32;                   // float step per k-chunk (a Psw k-chunk spans 2048)

    for (int iter = 0; iter <= N_ITER; ++iter) {
        // ---- zk = s @ P, K = 512 in chunks of 4, pipelined 1 chunk ahead -
        v8f acc0 = {}, acc1 = {}, acc2 = {}, acc3 = {};
        v2f a_c  = *(const v2f*)(arow);
        v2f b0_c = *(const v2f*)(pb0);
        v2f b1_c = *(const v2f*)(pb1);
        v2f b2_c = *(const v2f*)(pb2);
        v2f b3_c = *(const v2f*)(pb3);
        #pragma unroll 4
        for (int kc = 0; kc < DPAD / 4 - 1; ++kc) {
            const size_t off = (size_t)(kc + 1) * kcstep;
            v2f a_n  = *(const v2f*)(arow + (kc + 1) * 4);
            v2f b0_n = *(const v2f*)(pb0 + off);
            v2f b1_n = *(const v2f*)(pb1 + off);
            v2f b2_n = *(const v2f*)(pb2 + off);
            v2f b3_n = *(const v2f*)(pb3 + off);
            acc0 = __builtin_amdgcn_wmma_f32_16x16x4_f32(false, a_c, false, b0_c, (short)0, acc0, false, false);
            acc1 = __builtin_amdgcn_wmma_f32_16x16x4_f32(false, a_c, false, b1_c, (short)0, acc1, false, false);
            acc2 = __builtin_amdgcn_wmma_f32_16x16x4_f32(false, a_c, false, b2_c, (short)0, acc2, false, false);
            acc3 = __builtin_amdgcn_wmma_f32_16x16x4_f32(false, a_c, false, b3_c, (short)0, acc3, false, false);
            a_c = a_n; b0_c = b0_n; b1_c = b1_n; b2_c = b2_n; b3_c = b3_n;
        }
        acc0 = __builtin_amdgcn_wmma_f32_16x16x4_f32(false, a_c, false, b0_c, (short)0, acc0, false, false);
        acc1 = __builtin_amdgcn_wmma_f32_16x16x4_f32(false, a_c, false, b1_c, (short)0, acc1, false, false);
        acc2 = __builtin_amdgcn_wmma_f32_16x16x4_f32(false, a_c, false, b2_c, (short)0, acc2, false, false);
        acc3 = __builtin_amdgcn_wmma_f32_16x16x4_f32(false, a_c, false, b3_c, (short)0, acc3, false, false);

        v8f acc[4] = { acc0, acc1, acc2, acc3 };

        // ---- epilogue: zk = acc + q; toproj ------------------------------
        v8f zk[4], tp[4], sf[4];
        #pragma unroll
        for (int t = 0; t < 4; ++t) {
            const int col = (wave + t * 8) * 16 + lm;
            #pragma unroll
            for (int v = 0; v < 8; ++v) {
                const int row = v + hi * 8;
                const float qv = s_q[row * DPAD + col];
                const float sv = s_s[row * DPAD + col];
                const float yv = s_y[row * DPAD + col];
                const float z  = acc[t][v] + qv;
                zk[t][v] = z;
                sf[t][v] = sv;
                float p = (2.0f * z - sv - TWO_SIG * yv) * INV_12S;
                if (col >= DREAL) p = 0.0f;
                tp[t][v] = p;
            }
        }

        if (iter == N_ITER) {
            // final projection output: zk columns < 500
            #pragma unroll
            for (int t = 0; t < 4; ++t) {
                const int col = (wave + t * 8) * 16 + lm;
                if (col < DREAL) {
                    #pragma unroll
                    for (int v = 0; v < 8; ++v) {
                        const int row = rowblk + v + hi * 8;
                        out[(size_t)row * DREAL + col] = zk[t][v];
                    }
                }
            }
            return;
        }

        // ---- SOC row norms: u = cols 250..498, t = col 499 ---------------
        #pragma unroll
        for (int t = 0; t < 4; ++t) {
            const int c0 = (wave + t * 8) * 16;
            if (c0 > 498 || c0 + 15 < MDIM) continue;
            const int col  = c0 + lm;
            const bool inu = (col >= MDIM && col <= 498);
            #pragma unroll
            for (int v = 0; v < 8; ++v) {
                float u = inu ? tp[t][v] * tp[t][v] : 0.0f;
                u += __shfl_xor(u, 1, 32);
                u += __shfl_xor(u, 2, 32);
                u += __shfl_xor(u, 4, 32);
                u += __shfl_xor(u, 8, 32);
                if (lm == 0) atomicAdd(&s_norm[v + hi * 8], u);
                if (col == 499) s_t[v + hi * 8] = tp[t][v];
            }
        }
        __syncthreads();

        if (tid < 16) {
            const float nu = sqrtf(s_norm[tid]);
            const float tt = s_t[tid];
            float us, tn;
            if (nu <= tt)       { us = 1.0f; tn = tt; }
            else if (nu <= -tt) { us = 0.0f; tn = 0.0f; }
            else { const float f = 0.5f * (tt + nu); us = f / (nu + 1e-12f); tn = f; }
            s_us[tid] = us;
            s_tn[tid] = tn;
            s_norm[tid] = 0.0f;   // reset for next iteration
        }
        __syncthreads();

        // ---- s <- s + omega * (tk - zk) ----------------------------------
        #pragma unroll
        for (int t = 0; t < 4; ++t) {
            const int col = (wave + t * 8) * 16 + lm;
            #pragma unroll
            for (int v = 0; v < 8; ++v) {
                const int row = v + hi * 8;
                const float p = tp[t][v];
                float tk;
                if (col < MDIM)       tk = p;
                else if (col <= 498)  tk = p * s_us[row];
                else if (col == 499)  tk = s_tn[row];
                else                  tk = 0.0f;
                float sn = sf[t][v] + OMEGA_C * (tk - zk[t][v]);
                if (col >= DREAL) sn = 0.0f;
                s_s[row * DPAD + col] = sn;
            }
        }
        __syncthreads();
    }
}

extern "C" void kernel_launch(void* const* d_in, const int* in_sizes, int n_in,
                              void* d_out, int out_size, void* d_ws, size_t ws_size,
                              hipStream_t stream)
{
    const float* b    = (const float*)d_in[0];  // 1024x250
    const float* c    = (const float*)d_in[1];  // 1024x250
    const float* W1   = (const float*)d_in[2];  // 500x512
    const float* b1   = (const float*)d_in[3];  // 512
    const float* W2   = (const float*)d_in[4];  // 512x512
    const float* b2   = (const float*)d_in[5];  // 512
    const float* W3   = (const float*)d_in[6];  // 512x500
    const float* b3   = (const float*)d_in[7];  // 500
    const float* Aaug = (const float*)d_in[8];  // 250x500
    const float* Ainv = (const float*)d_in[9];  // 500x250
    float* out = (float*)d_out;

    float* ws = (float*)d_ws;
    const size_t S = (size_t)BATCH * DPAD;       // 524288 floats
    float* buf0 = ws;                            // x, then h2
    float* buf1 = ws + S;                        // h1, then y
    float* qbuf = ws + 2 * S;                    // q (1024x512)
    float* psw  = ws + 3 * S;                    // swizzled P (512x512)
    float* ptmp = ws + 3 * S + DPAD * DPAD;      // P row-major (512x512)

    // x = concat(bv, cv), padded
    pack_x_kernel<<<(BATCH * DPAD + 255) / 256, 256, 0, stream>>>(b, c, buf0);

    // h1 = relu(x @ W1 + b1)
    gemm_wmma_generic<<<64, 256, 0, stream>>>(buf0, DPAD, 1, BATCH, DREAL,
                                              W1, 512, 1, 512, b1, buf1, BATCH, 0);
    // h2 = relu(h1 @ W2 + b2)
    gemm_wmma_generic<<<64, 256, 0, stream>>>(buf1, DPAD, 1, BATCH, 512,
                                              W2, 512, 1, 512, b2, buf0, BATCH, 0);
    // y = h2 @ W3 + b3
    gemm_wmma_generic<<<64, 256, 0, stream>>>(buf0, DPAD, 1, BATCH, 512,
                                              W3, 500, 1, 500, b3, buf1, BATCH, 1);
    // q = bv @ Ainv^T   (W[k][n] = Ainv[n*250 + k])
    gemm_wmma_generic<<<64, 256, 0, stream>>>(b, MDIM, 1, BATCH, MDIM,
                                              Ainv, 1, MDIM, DREAL, nullptr, qbuf, BATCH, 3);
    // P = I - Aaug^T @ Ainv^T   (A[m][k] = Aaug[k*500 + m])
    gemm_wmma_generic<<<32, 256, 0, stream>>>(Aaug, 1, DREAL, DREAL, MDIM,
                                              Ainv, 1, MDIM, DREAL, nullptr, ptmp, DREAL, 2);
    // Pre-swizzle P into WMMA B-fragment order
    swizzle_pmat_kernel<<<(DPAD * DPAD + 255) / 256, 256, 0, stream>>>(ptmp, psw);

    // 1000-iteration scan + final projection, 16 rows per workgroup
    solver_kernel<<<BATCH / 16, 256, 0, stream>>>(psw, qbuf, buf1, out);
}